// SNN_53858889892454
// MI455X (gfx1250) — compile-verified
//
#include <hip/hip_runtime.h>

typedef __attribute__((ext_vector_type(2))) float v2f;
typedef __attribute__((ext_vector_type(8))) float v8f;

namespace {
constexpr int NIN     = 250;   // K of layer-1 GEMM
constexpr int KP      = 252;   // K padded to multiple of 4 (WMMA K=4)
constexpr int NH      = 128;   // hidden
constexpr int NOUT    = 5;
constexpr int NSTEPS  = 25;
constexpr int TILE_M  = 128;   // rows per block
constexpr int THREADS = 256;   // 8 wave32 waves
constexpr float BETA = 0.9f;
constexpr float THR  = 1.0f;
// LDS: x tile [128][252] + w1 pair-interleaved [(252/2)][128][2]
constexpr size_t SMEM_BYTES = (size_t)(TILE_M * KP + (KP / 2) * 2 * NH) * sizeof(float);
}

__global__ __launch_bounds__(THREADS)
void snn_fused_kernel(const float* __restrict__ x,
                      const float* __restrict__ w1,
                      const float* __restrict__ b1,
                      const float* __restrict__ w2,
                      const float* __restrict__ b2,
                      float* __restrict__ out)
{
    extern __shared__ float smem[];
    float* xs  = smem;                 // [TILE_M][KP], rows padded with zeros at k=250,251
    float* w1p = smem + TILE_M * KP;   // [(KP/2)][NH][2] : (k/2, h, k&1) <- w1[h][k]

    const int tid = threadIdx.x;

    // ---------------- Stage operands into LDS (coalesced global reads) ----------------
    {
        const float* xg = x + (size_t)blockIdx.x * TILE_M * NIN;
        for (int idx = tid; idx < TILE_M * NIN; idx += THREADS) {
            int m = idx / NIN;
            int k = idx - m * NIN;
            xs[m * KP + k] = xg[idx];
        }
        // zero K-pad of x tile (ensures padded K contributes exactly 0)
        for (int idx = tid; idx < TILE_M * 2; idx += THREADS)
            xs[(idx >> 1) * KP + NIN + (idx & 1)] = 0.0f;

        // w1 is [NH][NIN]; store transposed, pair-interleaved along k
        for (int idx = tid; idx < NH * NIN; idx += THREADS) {
            int h = idx / NIN;
            int k = idx - h * NIN;
            w1p[(k >> 1) * (2 * NH) + 2 * h + (k & 1)] = w1[idx];
        }
        // zero K-pad rows k=250,251 of w1p (avoid NaN*0 from garbage LDS)
        for (int idx = tid; idx < NH * 2; idx += THREADS)
            w1p[(NIN >> 1) * (2 * NH) + idx] = 0.0f;
    }
    __syncthreads();

    const int lane = tid & 31;
    const int lo   = lane & 15;   // N / M-row selector within 16
    const int hi   = lane >> 4;   // half-wave
    const int wave = tid >> 5;    // M-tile 0..7 (rows 16*wave .. 16*wave+15)

    // ---------------- Layer-1 GEMM: cur1 = x @ w1^T + b1 via V_WMMA_F32_16X16X4_F32 ---
    // acc[t] holds the 16x16 C/D tile for H-tile t. C/D layout: lane l, VGPR j ->
    //   row m = j + 8*hi (within M-tile), col h = 16*t + lo.
    v8f acc[8];
#pragma unroll
    for (int t = 0; t < 8; ++t) {
        float bv = b1[16 * t + lo];     // fold bias into accumulator init
#pragma unroll
        for (int j = 0; j < 8; ++j) acc[t][j] = bv;
    }

    // A fragment (f32 16x4): lane l -> M = l%16; VGPR0 = A[M][k0+2*hi], VGPR1 = A[M][k0+1+2*hi]
    const float* xrow = xs + (16 * wave + lo) * KP + 2 * hi;

    for (int k0 = 0; k0 < KP; k0 += 4) {
        v2f a = *(const v2f*)(xrow + k0);   // one ds_load_b64, reused for 8 WMMAs
        // B fragment (f32 4x16, mirrored layout): lane l -> N = l%16;
        // VGPR0 = B[k0+2*hi][N], VGPR1 = B[k0+1+2*hi][N] -> contiguous pair in w1p.
        const float* bp = w1p + ((k0 >> 1) + hi) * (2 * NH) + 2 * lo;
#pragma unroll
        for (int t = 0; t < 8; ++t) {
            v2f b = *(const v2f*)(bp + 32 * t);
            acc[t] = __builtin_amdgcn_wmma_f32_16x16x4_f32(
                false, a, false, b, (short)0, acc[t], false, false);
        }
    }

    // ---------------- 25-step LIF recurrence, fully in registers ----------------------
    // Per lane: 8 rows (m = j + 8*hi) x 8 hidden columns (h = 16*t + lo).
    float w2r[NOUT][8];
#pragma unroll
    for (int o = 0; o < NOUT; ++o)
#pragma unroll
        for (int t = 0; t < 8; ++t) w2r[o][t] = w2[o * NH + 16 * t + lo];
    float b2r[NOUT];
#pragma unroll
    for (int o = 0; o < NOUT; ++o) b2r[o] = b2[o];

    float mem1[8][8] = {};       // [t][j]
    float mem2[8][NOUT] = {};    // [j][o] (replicated across the 16 lanes of a half)
    float outa[8][NOUT] = {};

#pragma unroll 1
    for (int s = 0; s < NSTEPS; ++s) {
        float p[8][NOUT];        // partial cur2 over this lane's 8 hidden columns
#pragma unroll
        for (int j = 0; j < 8; ++j)
#pragma unroll
            for (int o = 0; o < NOUT; ++o) p[j][o] = 0.0f;

#pragma unroll
        for (int t = 0; t < 8; ++t) {
#pragma unroll
            for (int j = 0; j < 8; ++j) {
                float m = mem1[t][j];
                float reset = (m > THR) ? THR : 0.0f;        // from previous membrane
                m = BETA * m + acc[t][j] - reset;            // acc = static cur1
                mem1[t][j] = m;
                float sp = (m > THR) ? 1.0f : 0.0f;          // spike(m - THR)
#pragma unroll
                for (int o = 0; o < NOUT; ++o)
                    p[j][o] = fmaf(sp, w2r[o][t], p[j][o]);
            }
        }

        // butterfly-reduce partials across the 16 lanes that share each row
#pragma unroll
        for (int mask = 1; mask <= 8; mask <<= 1) {
#pragma unroll
            for (int j = 0; j < 8; ++j)
#pragma unroll
                for (int o = 0; o < NOUT; ++o)
                    p[j][o] += __shfl_xor(p[j][o], mask, 32);
        }

        // layer-2 LIF + spike-count accumulation
#pragma unroll
        for (int j = 0; j < 8; ++j) {
#pragma unroll
            for (int o = 0; o < NOUT; ++o) {
                float c2 = p[j][o] + b2r[o];
                float m2 = mem2[j][o];
                float r2 = (m2 > THR) ? THR : 0.0f;
                m2 = BETA * m2 + c2 - r2;
                mem2[j][o] = m2;
                outa[j][o] += (m2 > THR) ? 1.0f : 0.0f;
            }
        }
    }

    // ---------------- Epilogue: one lane per 16-group writes its 8 rows --------------
    if (lo == 0) {
        size_t row0 = (size_t)blockIdx.x * TILE_M + 16 * wave + 8 * hi;
#pragma unroll
        for (int j = 0; j < 8; ++j) {
            float* op = out + (row0 + j) * NOUT;
#pragma unroll
            for (int o = 0; o < NOUT; ++o) op[o] = outa[j][o];
        }
    }
}

extern "C" void kernel_launch(void* const* d_in, const int* in_sizes, int n_in,
                              void* d_out, int out_size, void* d_ws, size_t ws_size,
                              hipStream_t stream) {
    const float* x  = (const float*)d_in[0];
    const float* w1 = (const float*)d_in[1];
    const float* b1 = (const float*)d_in[2];
    const float* w2 = (const float*)d_in[3];
    const float* b2 = (const float*)d_in[4];
    float* out = (float*)d_out;

    const int B = in_sizes[0] / NIN;        // 131072
    const int grid = B / TILE_M;            // 1024 blocks

    snn_fused_kernel<<<grid, THREADS, SMEM_BYTES, stream>>>(x, w1, b1, w2, b2, out);
}